// TemporalFusionBlock_82222853915143
// MI455X (gfx1250) — compile-verified
//
#include <hip/hip_runtime.h>
#include <hip/hip_bf16.h>
#include <math.h>

typedef float v2f __attribute__((ext_vector_type(2)));
typedef float v8f __attribute__((ext_vector_type(8)));

#define BATCH 64
#define LOUT  256
#define CTOT  448
#define OCH   128
#define NTOT  (BATCH * LOUT)   // 16384

// workspace layout (float offsets)
#define FUSED_OFF 0u
#define WQ_OFF    (CTOT * NTOT)                // 7,340,032
#define Y_OFF     (WQ_OFF + OCH * CTOT)        // 7,397,376
#define MEAN_OFF  (Y_OFF + OCH * NTOT)         // 9,494,528
#define INV_OFF   (MEAN_OFF + OCH)             // 9,494,656

// ---- gfx1250 async global->LDS support (guarded; falls back to plain loads)
#if __has_builtin(__builtin_amdgcn_global_load_async_to_lds_b128) && \
    __has_builtin(__builtin_amdgcn_global_load_async_to_lds_b32)
#define HAVE_ASYNC_LDS 1
// Parameter types per hipcc diagnostics:
//   b128: 'int __attribute__((vector_size(16))) __device__ *'
//   b32 : 'int __device__ *'
typedef int v4i __attribute__((vector_size(4 * sizeof(int))));
typedef __attribute__((address_space(1))) v4i* gptr_v4i;
typedef __attribute__((address_space(3))) v4i* lptr_v4i;
typedef __attribute__((address_space(1))) int* gptr_i32;
typedef __attribute__((address_space(3))) int* lptr_i32;

__device__ __forceinline__ void async_wait_all() {
#if __has_builtin(__builtin_amdgcn_s_wait_asynccnt)
    __builtin_amdgcn_s_wait_asynccnt(0);
#else
    asm volatile("s_wait_asynccnt 0x0" ::: "memory");
#endif
}
#endif

// ---------------------------------------------------------------------------
// Kernel 1: adaptive avg-pool (torch semantics) + softmax(alpha) scale + concat
// One block per (b, c) row of the fused tensor. Row staged in LDS via
// GLOBAL_LOAD_ASYNC_TO_LDS_B128/B32 when available.
// fused layout: [c][b*256 + l]  (K-major, ready as GEMM B-matrix)
// ---------------------------------------------------------------------------
__global__ void pool_concat_kernel(const float* __restrict__ s1,
                                   const float* __restrict__ s2,
                                   const float* __restrict__ s3,
                                   const float* __restrict__ alpha,
                                   float* __restrict__ fused) {
    __shared__ float row[5000];
    const int idx = blockIdx.x;
    const int b = idx / CTOT;
    const int c = idx - b * CTOT;

    // softmax over the 3 fusion weights (redundant per block, negligible)
    const float a0 = alpha[0], a1 = alpha[1], a2 = alpha[2];
    const float mx = fmaxf(a0, fmaxf(a1, a2));
    const float e0 = expf(a0 - mx), e1 = expf(a1 - mx), e2 = expf(a2 - mx);
    const float inv = 1.0f / (e0 + e1 + e2);

    const float* src;
    int Lin;
    float w;
    if (c < 256)      { src = s1 + ((size_t)b * 256 + c)         * 5000; Lin = 5000; w = e0 * inv; }
    else if (c < 384) { src = s2 + ((size_t)b * 128 + (c - 256)) * 2500; Lin = 2500; w = e1 * inv; }
    else              { src = s3 + ((size_t)b * 64  + (c - 384)) * 625;  Lin = 625;  w = e2 * inv; }

#ifdef HAVE_ASYNC_LDS
    if ((Lin & 3) == 0) {               // stage1/stage2 rows: 16B aligned, len%4==0
        const int quads = Lin >> 2;
        for (int q = threadIdx.x; q < quads; q += 256)
            __builtin_amdgcn_global_load_async_to_lds_b128(
                (gptr_v4i)(src + 4 * q), (lptr_v4i)&row[4 * q], 0, 0);
    } else {                            // stage3 rows: only 4B aligned
        for (int i = threadIdx.x; i < Lin; i += 256)
            __builtin_amdgcn_global_load_async_to_lds_b32(
                (gptr_i32)(src + i), (lptr_i32)&row[i], 0, 0);
    }
    async_wait_all();
#else
    for (int i = threadIdx.x; i < Lin; i += 256) row[i] = src[i];
#endif
    __syncthreads();

    const int i = threadIdx.x;                     // 256 output positions
    const int start = (i * Lin) >> 8;              // floor(i*L/256)
    const int end   = ((i + 1) * Lin + 255) >> 8;  // ceil((i+1)*L/256)
    float s = 0.0f;
    for (int j = start; j < end; ++j) s += row[j];
    fused[(size_t)c * NTOT + b * LOUT + i] = s * w / (float)(end - start);
}

// ---------------------------------------------------------------------------
// Kernel 2: 2-bit symmetric weight quantization, per-output-channel scale.
// qp=+1, qn=-2; scale = max(|w|)/1 clamped at 1e-8; RNE rounding (rintf).
// ---------------------------------------------------------------------------
__global__ void quantize_kernel(const float* __restrict__ wgt,
                                float* __restrict__ wq) {
    __shared__ float red[256];
    const int o = blockIdx.x;
    const float* wr = wgt + (size_t)o * CTOT;
    float m = 0.0f;
    for (int i = threadIdx.x; i < CTOT; i += 256) m = fmaxf(m, fabsf(wr[i]));
    red[threadIdx.x] = m;
    __syncthreads();
    for (int s = 128; s > 0; s >>= 1) {
        if (threadIdx.x < s) red[threadIdx.x] = fmaxf(red[threadIdx.x], red[threadIdx.x + s]);
        __syncthreads();
    }
    const float scale = fmaxf(red[0], 1e-8f);
    const float rs = 1.0f / scale;
    for (int i = threadIdx.x; i < CTOT; i += 256) {
        float q = rintf(wr[i] * rs);
        q = fminf(fmaxf(q, -2.0f), 1.0f);
        wq[(size_t)o * CTOT + i] = q * scale;
    }
}

// ---------------------------------------------------------------------------
// Kernel 3: y[o][n] = sum_c wq[o][c] * fused[c][n] + bias[o]
// M=128, N=16384, K=448 via V_WMMA_F32_16X16X4_F32.
// Block = 256 threads (8 waves), covers M=128 x N=32; B-panel staged in LDS
// (async B128 when available). Fragment layouts per ISA 7.12.2:
// lanes 0-15 hold K pair {k,k+1}, lanes 16-31 hold {k+2,k+3};
// C/D vgpr v -> row m0+v (lanes 0-15) / m0+v+8 (lanes 16-31).
// ---------------------------------------------------------------------------
__global__ void wmma_gemm_kernel(const float* __restrict__ wq,
                                 const float* __restrict__ fused,
                                 const float* __restrict__ bias,
                                 float* __restrict__ y) {
    __shared__ float Bt[CTOT * 32];            // 57,344 bytes
    const int n0 = blockIdx.x * 32;

#ifdef HAVE_ASYNC_LDS
    // 448 rows x 32 floats = 3584 16-byte quads; all addresses 16B aligned
    for (int q = threadIdx.x; q < CTOT * 8; q += 256) {
        const int r = q >> 3, qc = q & 7;
        __builtin_amdgcn_global_load_async_to_lds_b128(
            (gptr_v4i)(fused + (size_t)r * NTOT + n0 + 4 * qc),
            (lptr_v4i)&Bt[r * 32 + 4 * qc], 0, 0);
    }
    async_wait_all();
#else
    for (int idx = threadIdx.x; idx < CTOT * 32; idx += 256) {
        const int r = idx >> 5, cn = idx & 31;
        Bt[idx] = fused[(size_t)r * NTOT + n0 + cn];
    }
#endif
    __syncthreads();

    const int lane  = threadIdx.x & 31;
    const int wid   = threadIdx.x >> 5;
    const int m0    = wid * 16;
    const int mrow  = m0 + (lane & 15);
    const int khalf = (lane >> 4) * 2;         // 0 for lanes 0-15, 2 for 16-31
    const int col   = lane & 15;

    v8f acc0 = {};
    v8f acc1 = {};
    const float* arow = wq + (size_t)mrow * CTOT;

    for (int k = 0; k < CTOT; k += 4) {
        const int kb = k + khalf;
        v2f a;  a.x  = arow[kb];               a.y  = arow[kb + 1];
        v2f b0; b0.x = Bt[kb * 32 + col];      b0.y = Bt[(kb + 1) * 32 + col];
        v2f b1; b1.x = Bt[kb * 32 + 16 + col]; b1.y = Bt[(kb + 1) * 32 + 16 + col];
        acc0 = __builtin_amdgcn_wmma_f32_16x16x4_f32(false, a, false, b0,
                                                     (short)0, acc0, false, false);
        acc1 = __builtin_amdgcn_wmma_f32_16x16x4_f32(false, a, false, b1,
                                                     (short)0, acc1, false, false);
    }

    const int rbase = m0 + 8 * (lane >> 4);
#pragma unroll
    for (int v = 0; v < 8; ++v) {
        const int row = rbase + v;
        const float bz = bias[row];
        y[(size_t)row * NTOT + n0 + col]      = acc0[v] + bz;
        y[(size_t)row * NTOT + n0 + 16 + col] = acc1[v] + bz;
    }
}

// ---------------------------------------------------------------------------
// Kernel 4: per-channel batch stats over (B, L) -> mean, 1/sqrt(var+eps)
// ---------------------------------------------------------------------------
__global__ void bn_stats_kernel(const float* __restrict__ y,
                                float* __restrict__ mean,
                                float* __restrict__ invstd) {
    __shared__ float rs[256];
    __shared__ float rss[256];
    const int o = blockIdx.x;
    float s = 0.0f, ss = 0.0f;
    for (int n = threadIdx.x; n < NTOT; n += 256) {
        const float v = y[(size_t)o * NTOT + n];
        s += v; ss += v * v;
    }
    rs[threadIdx.x] = s; rss[threadIdx.x] = ss;
    __syncthreads();
    for (int st = 128; st > 0; st >>= 1) {
        if (threadIdx.x < st) {
            rs[threadIdx.x]  += rs[threadIdx.x + st];
            rss[threadIdx.x] += rss[threadIdx.x + st];
        }
        __syncthreads();
    }
    if (threadIdx.x == 0) {
        const float m   = rs[0] * (1.0f / (float)NTOT);
        const float var = rss[0] * (1.0f / (float)NTOT) - m * m;
        mean[o]   = m;
        invstd[o] = rsqrtf(var + 1e-5f);
    }
}

// ---------------------------------------------------------------------------
// Kernel 5: BN affine + ELU + mean over L -> out (64, 128)
// One block per (b, o) pair; 256 threads = 256 length positions.
// ---------------------------------------------------------------------------
__global__ void bn_elu_mean_kernel(const float* __restrict__ y,
                                   const float* __restrict__ mean,
                                   const float* __restrict__ invstd,
                                   const float* __restrict__ gamma,
                                   const float* __restrict__ beta,
                                   float* __restrict__ out) {
    __shared__ float rs[256];
    const int idx = blockIdx.x;
    const int b = idx >> 7;        // / 128
    const int o = idx & 127;
    const float m = mean[o], is = invstd[o], g = gamma[o], bt = beta[o];

    const float v  = y[(size_t)o * NTOT + b * LOUT + threadIdx.x];
    const float yn = (v - m) * is * g + bt;
    const float act = yn > 0.0f ? yn : expm1f(yn);   // ELU(alpha=1)
    rs[threadIdx.x] = act;
    __syncthreads();
    for (int st = 128; st > 0; st >>= 1) {
        if (threadIdx.x < st) rs[threadIdx.x] += rs[threadIdx.x + st];
        __syncthreads();
    }
    if (threadIdx.x == 0) out[idx] = rs[0] * (1.0f / 256.0f);
}

// ---------------------------------------------------------------------------
extern "C" void kernel_launch(void* const* d_in, const int* in_sizes, int n_in,
                              void* d_out, int out_size, void* d_ws, size_t ws_size,
                              hipStream_t stream) {
    const float* stage1 = (const float*)d_in[0];
    const float* stage2 = (const float*)d_in[1];
    const float* stage3 = (const float*)d_in[2];
    const float* alpha  = (const float*)d_in[3];
    const float* weight = (const float*)d_in[4];
    const float* bias   = (const float*)d_in[5];
    const float* gamma  = (const float*)d_in[6];
    const float* beta   = (const float*)d_in[7];
    float* out = (float*)d_out;

    float* ws     = (float*)d_ws;
    float* fused  = ws + FUSED_OFF;
    float* wq     = ws + WQ_OFF;
    float* y      = ws + Y_OFF;
    float* meanb  = ws + MEAN_OFF;
    float* invb   = ws + INV_OFF;

    // 1) pooling + softmax scale + concat
    pool_concat_kernel<<<BATCH * CTOT, 256, 0, stream>>>(stage1, stage2, stage3,
                                                         alpha, fused);
    // 2) 2-bit weight quantization
    quantize_kernel<<<OCH, 256, 0, stream>>>(weight, wq);
    // 3) WMMA GEMM + bias
    wmma_gemm_kernel<<<NTOT / 32, 256, 0, stream>>>(wq, fused, bias, y);
    // 4) BN batch stats
    bn_stats_kernel<<<OCH, 256, 0, stream>>>(y, meanb, invb);
    // 5) BN + ELU + mean over L
    bn_elu_mean_kernel<<<BATCH * OCH, 256, 0, stream>>>(y, meanb, invb,
                                                        gamma, beta, out);
}